// chebyshev_nufft_48704929136780
// MI455X (gfx1250) — compile-verified
//
#include <hip/hip_runtime.h>
#include <hip/hip_bf16.h>
#include <math.h>
#include <stdint.h>

// ---------------------------------------------------------------------------
// Chebyshev NUFFT forward model on gfx1250.
// FFT replaced by centered DFT matrix F (256x256, symmetric, ortho-normalized)
// so both FFT stages become complex GEMMs on V_WMMA_F32_16X16X4_F32, with the
// shared B K-panels staged into LDS by the Tensor Data Mover (TDM).
// ---------------------------------------------------------------------------

typedef __attribute__((ext_vector_type(2))) float v2f;
typedef __attribute__((ext_vector_type(8))) float v8f;
typedef unsigned int v4u __attribute__((ext_vector_type(4)));
typedef int v8i_t __attribute__((ext_vector_type(8)));
typedef int v4i_t __attribute__((ext_vector_type(4)));

#define IMX   256
#define NPIX  (IMX * IMX)       // 65536
#define NCH   5
#define LTOT  (NCH * NCH)       // 25
#define NC    8
#define NB    (NC * LTOT)       // 200 batched complex images
#define KCHUNK 64               // K-panel depth staged in LDS per TDM transfer

// d_ws layout (float offsets). Needs ~211 MB.
#define OFF_FR    0
#define OFF_FI    (OFF_FR + NPIX)
#define OFF_BR    (OFF_FI + NPIX)            // 5*256 apodizer table (re)
#define OFF_BI    (OFF_BR + NCH * IMX)       // 5*256 apodizer table (im)
#define OFF_BUF0  262144                      // M planes, later K planes
#define PLANE     ((long)NB * NPIX)           // 13,107,200 floats per re/im plane
#define OFF_BUF1  (OFF_BUF0 + 2 * PLANE)      // T planes

// ---------------------------------------------------------------------------
// Table builders
// ---------------------------------------------------------------------------

// Centered ortho DFT matrix: F[u,x] = exp(-2*pi*i*(u-128)(x-128)/256)/16.
// fftshift(fft(ifftshift(.))) == multiplication by this matrix on both axes,
// and F is symmetric, so stage2 can reuse F directly (F^T == F).
__global__ void build_fmat_kernel(float* __restrict__ Fr, float* __restrict__ Fi) {
    int gid = blockIdx.x * blockDim.x + threadIdx.x;   // 65536 threads
    int u = gid >> 8, x = gid & 255;
    int ph = ((u - 128) * (x - 128)) & 255;            // exact phase mod 256
    float th = (float)ph * (float)(2.0 * M_PI / 256.0);
    Fr[gid] =  cosf(th) * (1.0f / 16.0f);
    Fi[gid] = -sinf(th) * (1.0f / 16.0f);
}

// Integer-order Bessel J_n(-pi/2), n = 0..4 (J_{-n} = (-1)^n J_n).
__device__ __forceinline__ float jbes(int k) {
    const float J[5] = {0.4720012f, -0.5668551f, 0.2497215f, -0.0690044f, 0.0138664f};
    int a = k < 0 ? -k : k;
    float v = J[a];
    if (k < 0 && (a & 1)) v = -v;
    return v;
}

// b[p][x] = sum_q A[p,q] * T_q(2*r_x), A[p,q] = 4 i^q J_{(q+p)/2} J_{(q-p)/2} * s_p s_q,
// nonzero only for (p-q) even; s_0 = 0.5. Both image dims identical -> one table.
__global__ void build_btab_kernel(float* __restrict__ br, float* __restrict__ bi) {
    int gid = blockIdx.x * blockDim.x + threadIdx.x;   // 5*256 threads
    if (gid >= NCH * IMX) return;
    int p = gid >> 8;
    int x = gid & 255;
    float r = (float)(x - 128) / 256.0f;
    float t = fminf(fmaxf(2.0f * r, -1.0f), 1.0f);
    float ang = acosf(t);
    float sp = (p == 0) ? 0.5f : 1.0f;
    float sr = 0.0f, si = 0.0f;
    for (int q = 0; q < NCH; ++q) {
        if (((p - q) & 1) != 0) continue;
        float mag = 4.0f * jbes((q + p) / 2) * jbes((q - p) / 2) * sp * ((q == 0) ? 0.5f : 1.0f);
        float Tq = cosf((float)q * ang);
        switch (q & 3) {                                // i^q factor
            case 0: sr += mag * Tq; break;
            case 1: si += mag * Tq; break;
            case 2: sr -= mag * Tq; break;
            case 3: si -= mag * Tq; break;
        }
    }
    br[gid] = sr;
    bi[gid] = si;
}

// ---------------------------------------------------------------------------
// Modulation: M[c,l] = img[c] * (b[p0](x) * b[p1](y)), split re/im planes.
// ---------------------------------------------------------------------------
__global__ void modulate_kernel(const float* __restrict__ img,   // interleaved complex (C,X,Y)
                                const float* __restrict__ br,
                                const float* __restrict__ bi,
                                float* __restrict__ Mr,
                                float* __restrict__ Mi) {
    long gid = (long)blockIdx.x * blockDim.x + threadIdx.x;      // NB * NPIX
    int pix = (int)(gid & 65535);
    int cl  = (int)(gid >> 16);
    int l = cl % LTOT, c = cl / LTOT;
    int x = pix >> 8, y = pix & 255;
    int p0 = l / NCH, p1 = l % NCH;
    float s0r = br[p0 * IMX + x], s0i = bi[p0 * IMX + x];
    float s1r = br[p1 * IMX + y], s1i = bi[p1 * IMX + y];
    float sr = s0r * s1r - s0i * s1i;
    float si = s0r * s1i + s0i * s1r;
    long ii = ((long)c * NPIX + pix) * 2;
    float vr = img[ii], vi = img[ii + 1];
    Mr[gid] = vr * sr - vi * si;
    Mi[gid] = vr * si + vi * sr;
}

// ---------------------------------------------------------------------------
// TDM: DMA a 64-row x 16-col f32 tile (row stride 256 floats) into LDS.
// D# per CDNA5 ISA:
//   group0: count=1 | lds_addr | global_addr | type=2
//   group1: data_size=4B, tensor_dim0=256, tensor_dim1=256,
//           tile_dim0=16, tile_dim1=64, tensor_dim0_stride=256
//   groups 2/3 (+spare group): zero (2-D tile)
// Toolchain uses the 6-arg builtin form:
//   (uint32x4 g0, int32x8 g1, int32x4 g2, int32x4 g3, int32x8 g4, i32 cpol)
// ---------------------------------------------------------------------------
__device__ __forceinline__ void tdm_load_tile_f32(unsigned ldsByteAddr, const float* gsrc) {
    unsigned long long ga = (unsigned long long)(uintptr_t)gsrc;
    v4u g0;
    g0.x = 1u;                                              // count=1 (valid descriptor)
    g0.y = ldsByteAddr;                                     // LDS byte address
    g0.z = (unsigned)(ga & 0xFFFFFFFFull);                  // global_addr[31:0]
    g0.w = (unsigned)((ga >> 32) & 0x01FFFFFFull) | (2u << 30); // global_addr[56:32], type=2
    v8i_t g1;
    g1[0] = 0x00020000;                                     // data_size=2 (4 bytes)
    g1[1] = (int)(256u << 16);                              // tensor_dim0 = 256 (low 16)
    g1[2] = (int)(256u << 16);                              // tensor_dim1 = 256 (low 16)
    g1[3] = (int)(16u  << 16);                              // tile_dim0 = 16
    g1[4] = KCHUNK;                                         // tile_dim1 = 64, tile_dim2 = 0
    g1[5] = 256;                                            // tensor_dim0_stride = 256
    g1[6] = 0;
    g1[7] = 0;
    v4i_t z4 = {0, 0, 0, 0};
    v8i_t z8 = {0, 0, 0, 0, 0, 0, 0, 0};
    __builtin_amdgcn_tensor_load_to_lds(g0, g1, z4, z4, z8, 0);
}

// ---------------------------------------------------------------------------
// Batched complex GEMM, C = A * B (256x256x256), via V_WMMA_F32_16X16X4_F32.
// Block = 8 waves computing a 128x16 output strip: wave w owns rows
// [tm0 + 16w, +16), all waves share one Kx16 B tile. The shared B K-panel
// (re+im) is staged into LDS by the TDM (wave 0 issues, TENSORcnt-synced),
// double-buffered so the DMA overlaps WMMA compute. A-fragments are
// contiguous global_load_b64 from L2, with prefetch of the next chunk.
// aStride/bStride == 0 selects a batch-shared operand (the DFT matrix).
// f32 WMMA has no A/B negation -> negate B-imag fragment in VALU for the
// (-ai*bi) term of the complex product.
// ---------------------------------------------------------------------------
__global__ void gemm_cplx_kernel(const float* __restrict__ Ar, const float* __restrict__ Ai, long aStride,
                                 const float* __restrict__ Br, const float* __restrict__ Bi, long bStride,
                                 float* __restrict__ Cr, float* __restrict__ Ci, long cStride) {
    __shared__ __align__(128) float ldsB[2][2][KCHUNK * 16];   // [buf][re/im][row*16+col]

    int lane = threadIdx.x & 31;
    int wave = threadIdx.x >> 5;
    int g = blockIdx.x;
    int batch = g >> 5;                 // 32 strips per image (2 row-strips x 16 col-tiles)
    int strip = g & 31;
    int tm = ((strip >> 4) << 7) + (wave << 4);   // this wave's 16 output rows
    int tn = (strip & 15) << 4;                   // block-shared 16 output cols

    const float* ar_p = Ar + (long)batch * aStride;
    const float* ai_p = Ai + (long)batch * aStride;
    const float* br_p = Br + (long)batch * bStride + tn;   // B tile column base
    const float* bi_p = Bi + (long)batch * bStride + tn;

    v8f dr = {};
    v8f di = {};

    // 16x4 f32 A fragment layout: lanes 0-15 -> M=lane, K in {k0,k0+1};
    // lanes 16-31 -> M=lane-16, K in {k0+2,k0+3}. B (4x16) mirrored on N.
    int rn = lane & 15;                 // row of A / col of B
    int kb = (lane >> 4) << 1;          // K sub-offset per half-wave
    const float* arow = ar_p + (long)(tm + rn) * 256;
    const float* airow = ai_p + (long)(tm + rn) * 256;

    unsigned ldsBase = (unsigned)(uintptr_t)(&ldsB[0][0][0]); // low 32 bits = LDS byte offset
    const unsigned planeBytes = KCHUNK * 16 * 4;              // 4 KB per plane

    // Prime buffer 0 with K-chunk 0 (wave 0 drives the TDM for the block).
    if (threadIdx.x < 32) {
        tdm_load_tile_f32(ldsBase + 0 * planeBytes, br_p);
        tdm_load_tile_f32(ldsBase + 1 * planeBytes, bi_p);
        __builtin_amdgcn_s_wait_tensorcnt(0);
    }
    __syncthreads();

    for (int c = 0; c < 256 / KCHUNK; ++c) {
        int cur = c & 1;
        int cbase = c * KCHUNK;

        // Kick off the DMA for the next K-chunk while this one computes.
        if (c + 1 < 256 / KCHUNK && threadIdx.x < 32) {
            unsigned nb = ldsBase + (unsigned)((c + 1) & 1) * 2 * planeBytes;
            tdm_load_tile_f32(nb + 0 * planeBytes, br_p + (long)(cbase + KCHUNK) * 256);
            tdm_load_tile_f32(nb + 1 * planeBytes, bi_p + (long)(cbase + KCHUNK) * 256);
        }
        // Prefetch next chunk of this wave's A rows into cache.
        if (c + 1 < 256 / KCHUNK) {
            __builtin_prefetch(arow + cbase + KCHUNK, 0, 3);
            __builtin_prefetch(airow + cbase + KCHUNK, 0, 3);
        }

        for (int kl = 0; kl < KCHUNK; kl += 4) {
            int ka = cbase + kl + kb;
            v2f a_r = *(const v2f*)(arow + ka);
            v2f a_i = *(const v2f*)(airow + ka);
            const float* lb_r = &ldsB[cur][0][(kl + kb) * 16 + rn];
            const float* lb_i = &ldsB[cur][1][(kl + kb) * 16 + rn];
            v2f b_r, b_i, nb_i;
            b_r.x = lb_r[0];
            b_r.y = lb_r[16];
            b_i.x = lb_i[0];
            b_i.y = lb_i[16];
            nb_i.x = -b_i.x;
            nb_i.y = -b_i.y;

            dr = __builtin_amdgcn_wmma_f32_16x16x4_f32(false, a_r, false, b_r,  (short)0, dr, false, false);
            dr = __builtin_amdgcn_wmma_f32_16x16x4_f32(false, a_i, false, nb_i, (short)0, dr, false, false);
            di = __builtin_amdgcn_wmma_f32_16x16x4_f32(false, a_r, false, b_i,  (short)0, di, false, false);
            di = __builtin_amdgcn_wmma_f32_16x16x4_f32(false, a_i, false, b_r,  (short)0, di, false, false);
        }

        __syncthreads();                                   // chunk's LDS reads complete
        if (c + 1 < 256 / KCHUNK && threadIdx.x < 32) {
            __builtin_amdgcn_s_wait_tensorcnt(0);          // next buffer landed
        }
        __syncthreads();
    }

    // 16x16 f32 D layout: VGPR r holds M = r (lanes 0-15) / M = r+8 (lanes 16-31).
    int n = tn + rn;
    int mhalf = (lane >> 4) * 8;
    float* cr_p = Cr + (long)batch * cStride;
    float* ci_p = Ci + (long)batch * cStride;
#pragma unroll
    for (int r = 0; r < 8; ++r) {
        int m = tm + mhalf + r;
        cr_p[(long)m * 256 + n] = dr[r];
        ci_p[(long)m * 256 + n] = di[r];
    }
}

// ---------------------------------------------------------------------------
// Gather + Chebyshev contraction. K grid (104 MB) stays L2-resident on MI455X
// (192 MB L2), so the 400 random loads per point mostly hit L2.
// ---------------------------------------------------------------------------
__global__ void gather_combine_kernel(const float* __restrict__ trj,
                                      const float* __restrict__ Kr,
                                      const float* __restrict__ Ki,
                                      float* __restrict__ out, int P) {
    int p = blockIdx.x * blockDim.x + threadIdx.x;
    if (p >= P) return;
    float tx = trj[2 * p], ty = trj[2 * p + 1];      // GRID_OS == 1
    float mx = rintf(tx), my = rintf(ty);            // round-half-even like np.round
    float dx = tx - mx, dy = ty - my;
    int ix = min(max((int)mx + 128, 0), 255);
    int iy = min(max((int)my + 128, 0), 255);
    int flat = ix * 256 + iy;
    float ax = acosf(fminf(fmaxf(2.0f * dx, -1.0f), 1.0f));
    float ay = acosf(fminf(fmaxf(2.0f * dy, -1.0f), 1.0f));
    float w0[NCH], w1[NCH];
#pragma unroll
    for (int q = 0; q < NCH; ++q) {
        w0[q] = cosf((float)q * ax);
        w1[q] = cosf((float)q * ay);
    }
#pragma unroll
    for (int c = 0; c < NC; ++c) {
        float accr = 0.0f, acci = 0.0f;
#pragma unroll
        for (int l = 0; l < LTOT; ++l) {
            float w = w0[l / NCH] * w1[l % NCH];
            long off = ((long)(c * LTOT + l)) * NPIX + flat;
            accr += w * Kr[off];
            acci += w * Ki[off];
        }
        long o = 2 * ((long)c * P + p);
        out[o]     = accr;
        out[o + 1] = acci;
    }
}

// ---------------------------------------------------------------------------
// Launch
// ---------------------------------------------------------------------------
extern "C" void kernel_launch(void* const* d_in, const int* in_sizes, int n_in,
                              void* d_out, int out_size, void* d_ws, size_t ws_size,
                              hipStream_t stream) {
    const float* img = (const float*)d_in[0];   // complex64 (1,8,256,256) interleaved
    const float* trj = (const float*)d_in[1];   // float32 (1,P,2)
    float* out = (float*)d_out;                 // complex64 (1,8,P) interleaved
    float* ws  = (float*)d_ws;
    int P = in_sizes[1] / 2;                    // 200000

    float* Fr = ws + OFF_FR;
    float* Fi = ws + OFF_FI;
    float* br = ws + OFF_BR;
    float* bi = ws + OFF_BI;
    float* Mr = ws + OFF_BUF0;                  // buf0: M planes, later K planes
    float* Mi = ws + OFF_BUF0 + PLANE;
    float* Tr = ws + OFF_BUF1;                  // buf1: T planes
    float* Ti = ws + OFF_BUF1 + PLANE;

    // 1. tables
    build_fmat_kernel<<<NPIX / 256, 256, 0, stream>>>(Fr, Fi);
    build_btab_kernel<<<(NCH * IMX + 255) / 256, 256, 0, stream>>>(br, bi);

    // 2. modulation: 200 complex images, re/im planes
    modulate_kernel<<<(int)(((long)NB * NPIX) / 256), 256, 0, stream>>>(img, br, bi, Mr, Mi);

    // 3. stage 1: T = F * M   (A shared, stride 0)
    gemm_cplx_kernel<<<NB * 32, 256, 0, stream>>>(Fr, Fi, 0L,
                                                  Mr, Mi, (long)NPIX,
                                                  Tr, Ti, (long)NPIX);

    // 4. stage 2: K = T * F   (B shared, stride 0), written over buf0
    gemm_cplx_kernel<<<NB * 32, 256, 0, stream>>>(Tr, Ti, (long)NPIX,
                                                  Fr, Fi, 0L,
                                                  Mr, Mi, (long)NPIX);

    // 5. gather + contraction
    gather_combine_kernel<<<(P + 255) / 256, 256, 0, stream>>>(trj, Mr, Mi, out, P);
}